// ModelNew_25056839204981
// MI455X (gfx1250) — compile-verified
//
#include <hip/hip_runtime.h>
#include <hip/hip_bf16.h>

// Problem constants (from the reference)
constexpr int CN   = 16;            // batch
constexpr int CIN  = 3;
constexpr int COUT = 16;
constexpr int KW3  = 3;
constexpr int DD   = 64, HH = 64, WW = 64;
constexpr int OD   = 62, OH = 62, OW = 62;           // VALID conv output dims
constexpr int KTOT = CIN * KW3 * KW3 * KW3;          // 81
constexpr int PIX   = CN * OH * OW;                  // 61504 output pixels (n,h,w)
constexpr int NTILES = PIX / 16;                     // 3844 (divides exactly)
constexpr int PLANE  = HH * WW;                      // 4096 elems per depth plane

// K dim permuted as (kd, cin, kh, kw); each kd-group padded 27 -> 28 = 7 chunks of 4.
// Total K' = 84 = 21 chunks. Group g's B fragments at depth d == group g-1's at d+1.

typedef __attribute__((ext_vector_type(2))) float v2f;
typedef __attribute__((ext_vector_type(8))) float v8f;

__device__ __forceinline__ v8f wmma4(v2f a, v2f b, v8f c) {
    return __builtin_amdgcn_wmma_f32_16x16x4_f32(false, a, false, b, (short)0, c,
                                                 false, false);
}

// One conv step over planes (d, d+1, d+2) held in windows (wa, wb, wc).
__device__ __forceinline__ v8f conv_step(const v2f (&af)[21],
                                         const v2f (&wa)[7],
                                         const v2f (&wb)[7],
                                         const v2f (&wc)[7]) {
    v8f acc = {};
#pragma unroll
    for (int c = 0; c < 7; ++c) acc = wmma4(af[c],      wa[c], acc);
#pragma unroll
    for (int c = 0; c < 7; ++c) acc = wmma4(af[7 + c],  wb[c], acc);
#pragma unroll
    for (int c = 0; c < 7; ++c) acc = wmma4(af[14 + c], wc[c], acc);
    return acc;
}

// Refill one kd-window from plane pointer xp (wave-uniform) + per-lane byte offsets.
__device__ __forceinline__ void load_plane(v2f (&wg)[7], const float* xp,
                                           const unsigned (&voff)[7][2]) {
#pragma unroll
    for (int c = 0; c < 7; ++c) {
        wg[c][0] = *(const float*)((const char*)xp + voff[c][0]);
        wg[c][1] = *(const float*)((const char*)xp + voff[c][1]);
    }
}

__device__ __forceinline__ void min_acc(v8f& mn, v8f acc) {
#pragma unroll
    for (int r = 0; r < 8; ++r) mn[r] = fminf(mn[r], acc[r]);
}

__global__ __launch_bounds__(256)
void conv3d_min_softmax_wmma(const float* __restrict__ x,
                             const float* __restrict__ wgt,
                             const float* __restrict__ bias,
                             float* __restrict__ out)
{
    const int wave = threadIdx.x >> 5;
    const int lane = threadIdx.x & 31;
    const int tile = blockIdx.x * 8 + wave;
    if (tile >= NTILES) return;                      // wave-uniform: EXEC stays all-1s

    const int col   = lane & 15;                     // GEMM N column (pixel) / A row (cout)
    const int khalf = lane >> 4;                     // which K pair this half-wave holds

    // ---- decode this lane's output pixel p = (n, h, w) ----
    const int p   = tile * 16 + col;
    const int n   = p / (OH * OW);
    const int rem = p - n * (OH * OW);
    const int h   = rem / OW;
    const int w   = rem - h * OW;

    // ---- per-lane B gather byte-offsets (kd excluded; one set reused for all planes) ----
    const unsigned nbase = (unsigned)n * (CIN * DD * HH * WW);
    unsigned voff[7][2];
#pragma unroll
    for (int c = 0; c < 7; ++c) {
#pragma unroll
        for (int j = 0; j < 2; ++j) {
            const int l = c * 4 + khalf * 2 + j;     // slot within kd-group: 0..27
            unsigned off = nbase;                    // safe in-bounds addr for pad slot
            if (l < 27) {
                const int cin = l / 9;
                const int r0  = l - cin * 9;
                const int kh  = r0 / 3;
                const int kw  = r0 - kh * 3;
                off = nbase + (unsigned)(cin * (DD * HH * WW) + (h + kh) * WW + (w + kw));
            }
            voff[c][j] = off * 4u;                   // bytes
        }
    }

    // ---- A fragments (weights) in permuted K order; pad slots get zero weight ----
    v2f af[21];
#pragma unroll
    for (int g = 0; g < 3; ++g) {                    // kd
#pragma unroll
        for (int c = 0; c < 7; ++c) {
#pragma unroll
            for (int j = 0; j < 2; ++j) {
                const int l = c * 4 + khalf * 2 + j;
                float wv = 0.0f;
                if (l < 27) {
                    const int cin = l / 9;
                    const int r0  = l - cin * 9;
                    const int kh  = r0 / 3;
                    const int kw  = r0 - kh * 3;
                    wv = wgt[col * KTOT + cin * 27 + g * 9 + kh * 3 + kw];
                }
                af[g * 7 + c][j] = wv;
            }
        }
    }

    // ---- sliding 3-plane window; prologue loads planes 0,1,2 ----
    v2f w0[7], w1[7], w2[7];
    const float* xp = x;                             // wave-uniform plane pointer
    load_plane(w0, xp, voff); xp += PLANE;
    load_plane(w1, xp, voff); xp += PLANE;
    load_plane(w2, xp, voff); xp += PLANE;           // xp -> plane 3

    v8f mn;
#pragma unroll
    for (int r = 0; r < 8; ++r) mn[r] = 3.4e38f;

    // 62 = 20*3 + 2 depth iterations; 3-phase rotation avoids register shuffles.
#pragma unroll 1
    for (int t = 0; t < 20; ++t) {
        { v8f a = conv_step(af, w0, w1, w2); min_acc(mn, a); load_plane(w0, xp, voff); xp += PLANE; }
        { v8f a = conv_step(af, w1, w2, w0); min_acc(mn, a); load_plane(w1, xp, voff); xp += PLANE; }
        { v8f a = conv_step(af, w2, w0, w1); min_acc(mn, a); load_plane(w2, xp, voff); xp += PLANE; }
    }
    { v8f a = conv_step(af, w0, w1, w2); min_acc(mn, a); load_plane(w0, xp, voff); } // d=60, plane 63
    { v8f a = conv_step(af, w1, w2, w0); min_acc(mn, a); }                           // d=61

    // ---- bias + softmax over the 16 channels of this lane's pixel ----
    // C/D layout: VGPR r holds cout = r + 8*khalf, pixel N = col.
    float v[8];
    float vmax = -3.4e38f;
#pragma unroll
    for (int r = 0; r < 8; ++r) {
        v[r] = mn[r] + bias[khalf * 8 + r];
        vmax = fmaxf(vmax, v[r]);
    }
    vmax = fmaxf(vmax, __shfl_xor(vmax, 16, 32));    // combine the two lane halves

    float vsum = 0.0f;
#pragma unroll
    for (int r = 0; r < 8; ++r) {
        v[r] = __expf(v[r] - vmax);
        vsum += v[r];
    }
    vsum += __shfl_xor(vsum, 16, 32);
    const float inv = 1.0f / vsum;

    // out[n][cout][h][w]
#pragma unroll
    for (int r = 0; r < 8; ++r) {
        const int cout = khalf * 8 + r;
        out[(((long)n * COUT + cout) * OH + h) * OW + w] = v[r] * inv;
    }
}

extern "C" void kernel_launch(void* const* d_in, const int* in_sizes, int n_in,
                              void* d_out, int out_size, void* d_ws, size_t ws_size,
                              hipStream_t stream)
{
    const float* x    = (const float*)d_in[0];   // [16,3,64,64,64]
    const float* wgt  = (const float*)d_in[1];   // [16,3,3,3,3]
    const float* bias = (const float*)d_in[2];   // [16]
    float* out        = (float*)d_out;           // [16,16,62,62]

    const int blocks = (NTILES + 7) / 8;         // 8 waves (tiles) per 256-thread block
    conv3d_min_softmax_wmma<<<blocks, 256, 0, stream>>>(x, wgt, bias, out);
}